// IouLabelPred_51977694216744
// MI455X (gfx1250) — compile-verified
//
#include <hip/hip_runtime.h>

#define RADIUS   5
#define NW       11          // 2*RADIUS+1
#define NOFF     121         // NW*NW
#define WDIM     72
#define WH       5184        // 72*72
#define NTHREADS 256

typedef __attribute__((ext_vector_type(4))) float v4f;
typedef __attribute__((address_space(1))) int* gint_p;   // global AS
typedef __attribute__((address_space(3))) int* lint_p;   // LDS AS

#if __has_builtin(__builtin_amdgcn_global_load_async_to_lds_b32)
#define ASYNC_GATHER 1
#else
#define ASYNC_GATHER 0
#endif

__global__ __launch_bounds__(NTHREADS)
void IouLabelPred_kernel(const float* __restrict__ feat,   // (B,4,72,72)
                         const int*   __restrict__ ind,    // (B)
                         const float* __restrict__ target, // (B,4)
                         float*       __restrict__ out)    // (B,72,72)
{
    __shared__ __align__(16) float val[WH];        // staged output tile
    __shared__ int   ic_s[NOFF];                   // scatter targets
    __shared__ __align__(16) float pred_s[NOFF*4]; // async-gathered preds

    const int b = blockIdx.x;
    const int t = threadIdx.x;

    // ---- Phase 1: fill staging tile with -1.0 (LDS b128 stores) ----
    v4f neg1 = {-1.0f, -1.0f, -1.0f, -1.0f};
    v4f* val4 = (v4f*)val;
    for (int i = t; i < WH/4; i += NTHREADS) val4[i] = neg1;

    // uniform per-block scalars (lower to s_loads)
    const int   idx = ind[b];
    const int   cw0 = idx % WDIM;
    const int   ch0 = idx / WDIM;
    const float t0 = target[4*b+0];
    const float t1 = target[4*b+1];
    const float t2 = target[4*b+2];
    const float t3 = target[4*b+3];

    int ic = 0, rw = 0, rh = 0;
    if (t < NOFF) {
        rw = t / NW - RADIUS;   // slow index (matches jnp.repeat)
        rh = t % NW - RADIUS;   // fast index (matches jnp.tile)
        int cw = cw0 + rw; cw = cw < 0 ? 0 : (cw > WDIM-1 ? WDIM-1 : cw);
        int ch = ch0 + rh; ch = ch < 0 ? 0 : (ch > WDIM-1 ? WDIM-1 : ch);
        ic = ch * WDIM + cw;
        ic_s[t] = ic;
        const float* gsrc = feat + (size_t)b * 4 * WH + ic;
#if ASYNC_GATHER
        // async DMA gather: 4 strided b32 copies global -> LDS (ASYNCcnt path)
        __builtin_amdgcn_global_load_async_to_lds_b32(
            (gint_p)(gsrc + 0*WH), (lint_p)(&pred_s[4*t + 0]), 0, 0);
        __builtin_amdgcn_global_load_async_to_lds_b32(
            (gint_p)(gsrc + 1*WH), (lint_p)(&pred_s[4*t + 1]), 0, 0);
        __builtin_amdgcn_global_load_async_to_lds_b32(
            (gint_p)(gsrc + 2*WH), (lint_p)(&pred_s[4*t + 2]), 0, 0);
        __builtin_amdgcn_global_load_async_to_lds_b32(
            (gint_p)(gsrc + 3*WH), (lint_p)(&pred_s[4*t + 3]), 0, 0);
#endif
    }
    __syncthreads();   // tile filled, ic_s visible

    if (t < NOFF) {
        // last-write-wins: offset t wins iff no later offset hits the same ic.
        // Duplicates only possible when the window clamps at a border.
        bool win = true;
        const bool interior = (cw0 >= RADIUS) && (cw0 + RADIUS < WDIM) &&
                              (ch0 >= RADIUS) && (ch0 + RADIUS < WDIM);
        if (!interior) {
            for (int j = t + 1; j < NOFF; ++j) win = win && (ic_s[j] != ic);
        }

        float pl, pr, pt, pb;
#if ASYNC_GATHER
#if __has_builtin(__builtin_amdgcn_s_wait_asynccnt)
        __builtin_amdgcn_s_wait_asynccnt(0);
#else
        asm volatile("s_wait_asynccnt 0" ::: "memory");
#endif
        pl = pred_s[4*t+0]; pr = pred_s[4*t+1];
        pt = pred_s[4*t+2]; pb = pred_s[4*t+3];
#else
        const float* gsrc = feat + (size_t)b * 4 * WH + ic;
        pl = gsrc[0*WH]; pr = gsrc[1*WH]; pt = gsrc[2*WH]; pb = gsrc[3*WH];
#endif
        const float tl = t0 + (float)rw;
        const float tr = t1 - (float)rw;
        const float tt = t2 + (float)rh;
        const float tb = t3 - (float)rh;
        const float inter = (fminf(pl, tl) + fminf(pr, tr)) *
                            (fminf(pb, tb) + fminf(pt, tt));
        const float uni   = (tl + tr) * (tt + tb) + (pl + pr) * (pt + pb) - inter;
        const float iou   = (inter + 1.0f) / (uni + 1.0f);
        if (win) val[ic] = iou;
    }
    __syncthreads();   // tile finalized

    // ---- Phase 3: stream tile to HBM, write-once, NT b128 stores ----
    float* dst  = out + (size_t)b * WH;
    v4f*   dst4 = (v4f*)dst;
    for (int i = t; i < WH/4; i += NTHREADS) {
        __builtin_nontemporal_store(val4[i], dst4 + i);
    }
}

extern "C" void kernel_launch(void* const* d_in, const int* in_sizes, int n_in,
                              void* d_out, int out_size, void* d_ws, size_t ws_size,
                              hipStream_t stream) {
    const float* feat   = (const float*)d_in[0];  // (128,8,4,72,72) f32
    const int*   ind    = (const int*)  d_in[1];  // (128,8,1) i32
    const float* target = (const float*)d_in[2];  // (128,8,4) f32
    float*       out    = (float*)d_out;          // (128,8,72,72) f32

    const int B = in_sizes[1];                    // 1024 batch elements
    dim3 grid(B), block(NTHREADS);
    hipLaunchKernelGGL(IouLabelPred_kernel, grid, block, 0, stream,
                       feat, ind, target, out);
}